// JointModel_27650999452046
// MI455X (gfx1250) — compile-verified
//
#include <hip/hip_runtime.h>

// ---------- problem constants (match reference) ----------
#define NGLOB 100000
#define NCLI  50000
#define ES_   1600000
#define EC_   800000

typedef __attribute__((ext_vector_type(2))) float v2f;
typedef __attribute__((ext_vector_type(8))) float v8f;

// ---------------- utility kernels ----------------
__global__ void fill_zero4_k(float4* __restrict__ p, unsigned n4) {
  unsigned i = blockIdx.x * blockDim.x + threadIdx.x;
  if (i < n4) p[i] = make_float4(0.f, 0.f, 0.f, 0.f);
}

__global__ void degree_k(const int* __restrict__ dst, float* __restrict__ cnt, int E) {
  int e = blockIdx.x * blockDim.x + threadIdx.x;
  if (e < E) atomicAdd(&cnt[dst[e]], 1.0f);
}

__global__ void invert_k(float* __restrict__ c, int n) {
  int i = blockIdx.x * blockDim.x + threadIdx.x;
  if (i < n) c[i] = 1.0f / fmaxf(c[i], 1.0f);
}

// Repack weight matrix [Krows x Ncols] row-major into pair-major:
// P[p*Ncols + n] = (W[2p][n], W[2p+1][n])  -> GEMM B-fragment = one b64 load.
__global__ void pack_pairs_k(const float* __restrict__ W, v2f* __restrict__ P,
                             int halfK, int Ncols) {
  int idx = blockIdx.x * blockDim.x + threadIdx.x;
  if (idx >= halfK * Ncols) return;
  int p = idx / Ncols, n = idx - p * Ncols;
  v2f v;
  v.x = W[(2 * p)     * Ncols + n];
  v.y = W[(2 * p + 1) * Ncols + n];
  P[idx] = v;
}

// scatter-add: one thread per (edge, 4-feature chunk).
// agg[dst[e], 4c:4c+4] += x[src[e], 4c:4c+4]
template<int CH, int LOG2CH>   // CH = DV/4 chunks per row
__global__ void scatter4_k(const float4* __restrict__ x, const int* __restrict__ src,
                           const int* __restrict__ dst, float* __restrict__ agg,
                           unsigned total) {
  unsigned idx = blockIdx.x * blockDim.x + threadIdx.x;
  if (idx >= total) return;
  unsigned e = idx >> LOG2CH;
  unsigned c = idx & (CH - 1);
  float4 v = x[(size_t)src[e] * CH + c];
  float* a = agg + ((size_t)dst[e] * CH + c) * 4;
  atomicAdd(a + 0, v.x);
  atomicAdd(a + 1, v.y);
  atomicAdd(a + 2, v.z);
  atomicAdd(a + 3, v.w);
}

// xcat[i, 0:64] = H[i, :];  xcat[i, 64:128] = S[ids[i], :]   (float4 chunks)
__global__ void concat4_k(const float4* __restrict__ H, const float4* __restrict__ S,
                          const int* __restrict__ ids, float4* __restrict__ xcat,
                          unsigned total) {
  unsigned idx = blockIdx.x * blockDim.x + threadIdx.x;
  if (idx >= total) return;
  unsigned i = idx >> 5;         // 32 float4 chunks per row of 128
  unsigned c = idx & 31u;
  xcat[idx] = (c < 16u) ? H[(size_t)i * 16 + c]
                        : S[(size_t)ids[i] * 16 + (c - 16u)];
}

// ---------------- fused SAGE GEMM (WMMA f32 16x16x4) ----------------
// out[m, 0:64] = (agg[m,:] * inv[m]) @ Wl + bl + X[m,:] @ Wr   (optionally ReLU)
// Wlp/Wrp are pair-packed (v2f per (k-pair, col)); one wave per 16-row tile.
template<int K, bool RELU>
__global__ __launch_bounds__(128)
void sage_gemm_k(const float* __restrict__ agg, const float* __restrict__ inv,
                 const float* __restrict__ X,
                 const v2f* __restrict__ Wlp, const float* __restrict__ bl,
                 const v2f* __restrict__ Wrp,
                 float* __restrict__ out, int numTiles) {
  const int wave = threadIdx.x >> 5;
  const int lane = threadIdx.x & 31;
  const int tile = blockIdx.x * 4 + wave;
  if (tile >= numTiles) return;

  const int hh  = lane >> 4;      // lane half selects K pair within the k-step
  const int lan = lane & 15;      // A: row index; B/C: column index
  const int m   = tile * 16 + lan;
  const float invm = inv[m];
  const float* __restrict__ Arow = agg + (size_t)m * K;
  const float* __restrict__ Xrow = X   + (size_t)m * K;

  v8f acc[4] = {};   // 16 rows x 64 cols (4 n-tiles of 16)

  // pass 1: mean-aggregated neighbors @ Wl
  for (int k0 = 0; k0 < K; k0 += 4) {
    const int ka = k0 + 2 * hh;        // A layout: VGPR0=K ka, VGPR1=K ka+1
    const int p  = (k0 >> 1) + hh;     // packed pair row
    v2f a;
    a.x = Arow[ka]     * invm;
    a.y = Arow[ka + 1] * invm;
    const v2f* __restrict__ B = Wlp + (size_t)p * 64 + lan;
#pragma unroll
    for (int nt = 0; nt < 4; ++nt) {
      v2f b = B[nt * 16];              // single global_load_b64
      acc[nt] = __builtin_amdgcn_wmma_f32_16x16x4_f32(
          false, a, false, b, (short)0, acc[nt], false, false);
    }
  }
  // pass 2: root features @ Wr
  for (int k0 = 0; k0 < K; k0 += 4) {
    const int ka = k0 + 2 * hh;
    const int p  = (k0 >> 1) + hh;
    v2f a;
    a.x = Xrow[ka];
    a.y = Xrow[ka + 1];
    const v2f* __restrict__ B = Wrp + (size_t)p * 64 + lan;
#pragma unroll
    for (int nt = 0; nt < 4; ++nt) {
      v2f b = B[nt * 16];
      acc[nt] = __builtin_amdgcn_wmma_f32_16x16x4_f32(
          false, a, false, b, (short)0, acc[nt], false, false);
    }
  }

  // bias (+ReLU) + store.  C/D layout: VGPR r -> row r + 8*hh, lane&15 -> col.
#pragma unroll
  for (int nt = 0; nt < 4; ++nt) {
    const float bb = bl[nt * 16 + lan];
#pragma unroll
    for (int r = 0; r < 8; ++r) {
      float v = acc[nt][r] + bb;
      if (RELU) v = fmaxf(v, 0.0f);
      out[(size_t)(tile * 16 + r + 8 * hh) * 64 + nt * 16 + lan] = v;
    }
  }
}

// ---------------- output head: logits = xcat @ W + b  (K=128, 16 cols) ----------------
__global__ __launch_bounds__(128)
void head_gemm_k(const float* __restrict__ X, const v2f* __restrict__ Wp,
                 const float* __restrict__ b, float* __restrict__ logits,
                 int numTiles) {
  const int wave = threadIdx.x >> 5;
  const int lane = threadIdx.x & 31;
  const int tile = blockIdx.x * 4 + wave;
  if (tile >= numTiles) return;

  const int hh  = lane >> 4;
  const int lan = lane & 15;
  const float* __restrict__ Xrow = X + (size_t)(tile * 16 + lan) * 128;

  v8f acc = {};
  for (int k0 = 0; k0 < 128; k0 += 4) {
    const int ka = k0 + 2 * hh;
    const int p  = (k0 >> 1) + hh;
    v2f a;
    a.x = Xrow[ka];
    a.y = Xrow[ka + 1];
    v2f bb = Wp[(size_t)p * 16 + lan];
    acc = __builtin_amdgcn_wmma_f32_16x16x4_f32(
        false, a, false, bb, (short)0, acc, false, false);
  }
  const float bias = b[lan];
#pragma unroll
  for (int r = 0; r < 8; ++r) {
    logits[(size_t)(tile * 16 + r + 8 * hh) * 16 + lan] = acc[r] + bias;
  }
}

// row-wise softmax over 16 classes, in place
__global__ void softmax16_k(float* __restrict__ p, int rows) {
  int i = blockIdx.x * blockDim.x + threadIdx.x;
  if (i >= rows) return;
  float* row = p + (size_t)i * 16;
  float v[16];
  float mx = -3.402823466e38f;
#pragma unroll
  for (int j = 0; j < 16; ++j) { v[j] = row[j]; mx = fmaxf(mx, v[j]); }
  float s = 0.0f;
#pragma unroll
  for (int j = 0; j < 16; ++j) { v[j] = __expf(v[j] - mx); s += v[j]; }
  const float is = 1.0f / s;
#pragma unroll
  for (int j = 0; j < 16; ++j) row[j] = v[j] * is;
}

// ---------------- host orchestration ----------------
static inline unsigned cdiv(unsigned a, unsigned b) { return (a + b - 1) / b; }

extern "C" void kernel_launch(void* const* d_in, const int* in_sizes, int n_in,
                              void* d_out, int out_size, void* d_ws, size_t ws_size,
                              hipStream_t stream) {
  (void)in_sizes; (void)n_in; (void)out_size; (void)ws_size;

  // ----- inputs (setup_inputs dict order, params depth-first as written) -----
  const float* x0   = (const float*)d_in[0];
  const float* x1   = (const float*)d_in[1];
  const float* S0   = (const float*)d_in[2];
  const int*   ids[2] = { (const int*)d_in[3], (const int*)d_in[4] };
  const int*   e0   = (const int*)d_in[5];
  const int*   e1   = (const int*)d_in[6];
  const int*   es   = (const int*)d_in[7];

  const float* Wl_s[2] = { (const float*)d_in[8],  (const float*)d_in[11] };
  const float* bl_s[2] = { (const float*)d_in[9],  (const float*)d_in[12] };
  const float* Wr_s[2] = { (const float*)d_in[10], (const float*)d_in[13] };
  const float* Wl_c[2][2] = { { (const float*)d_in[14], (const float*)d_in[17] },
                              { (const float*)d_in[22], (const float*)d_in[25] } };
  const float* bl_c[2][2] = { { (const float*)d_in[15], (const float*)d_in[18] },
                              { (const float*)d_in[23], (const float*)d_in[26] } };
  const float* Wr_c[2][2] = { { (const float*)d_in[16], (const float*)d_in[19] },
                              { (const float*)d_in[24], (const float*)d_in[27] } };
  const float* Wout[2] = { (const float*)d_in[20], (const float*)d_in[28] };
  const float* bout[2] = { (const float*)d_in[21], (const float*)d_in[29] };

  const int* src_s = es;        const int* dst_s = es + ES_;
  const int* src_c[2] = { e0, e1 };
  const int* dst_c[2] = { e0 + EC_, e1 + EC_ };

  // ----- workspace layout (floats) -----
  float* ws    = (float*)d_ws;
  float* agg   = ws;                       // 6,400,000  (N*64 == NC*128, reused)
  float* xcat  = agg  + 6400000;           // 6,400,000
  float* S1    = xcat + 6400000;           // 6,400,000
  float* Hbuf[2][2];
  Hbuf[0][0] = S1 + 6400000;               // client0 after layer0
  Hbuf[1][0] = Hbuf[0][0] + 3200000;       // client1 after layer0
  Hbuf[0][1] = Hbuf[1][0] + 3200000;       // client0 after layer1
  Hbuf[1][1] = Hbuf[0][1] + 3200000;       // client1 after layer1
  float* inv_s    = Hbuf[1][1] + 3200000;  // 100,000
  float* inv_c[2] = { inv_s + NGLOB, inv_s + NGLOB + NCLI };

  // pair-packed weights (v2f each): structure 4x(32*64), client 8x(64*64), head 2x(64*16)
  v2f* pk = (v2f*)(inv_c[1] + NCLI);
  v2f* Wl_sp[2] = { pk,                 pk + 2048 };          // 32*64
  v2f* Wr_sp[2] = { pk + 4096,          pk + 6144 };
  v2f* Wl_cp[2][2] = { { pk + 8192,  pk + 8192 + 4096 },      // 64*64 each
                       { pk + 8192 + 8192, pk + 8192 + 12288 } };
  v2f* Wr_cp[2][2] = { { pk + 24576, pk + 24576 + 4096 },
                       { pk + 24576 + 8192, pk + 24576 + 12288 } };
  v2f* Woutp[2] = { pk + 40960, pk + 40960 + 1024 };          // 64*16 each

  // ----- outputs -----
  float* S2   = (float*)d_out;                   // [N,64]
  float* outp[2];
  outp[0] = S2 + (size_t)NGLOB * 64;             // [NC,16]
  outp[1] = outp[0] + (size_t)NCLI * 16;         // [NC,16]

  const dim3 B256(256);

  // ===== pack all weight matrices into pair-major form =====
  for (int l = 0; l < 2; ++l) {
    pack_pairs_k<<<cdiv(32 * 64, 256), B256, 0, stream>>>(Wl_s[l], Wl_sp[l], 32, 64);
    pack_pairs_k<<<cdiv(32 * 64, 256), B256, 0, stream>>>(Wr_s[l], Wr_sp[l], 32, 64);
    for (int c = 0; c < 2; ++c) {
      pack_pairs_k<<<cdiv(64 * 64, 256), B256, 0, stream>>>(Wl_c[c][l], Wl_cp[c][l], 64, 64);
      pack_pairs_k<<<cdiv(64 * 64, 256), B256, 0, stream>>>(Wr_c[c][l], Wr_cp[c][l], 64, 64);
    }
  }
  for (int c = 0; c < 2; ++c)
    pack_pairs_k<<<cdiv(64 * 16, 256), B256, 0, stream>>>(Wout[c], Woutp[c], 64, 16);

  // ===== degree / inverse-degree (edges fixed -> once) =====
  fill_zero4_k<<<cdiv(NGLOB / 4, 256), B256, 0, stream>>>((float4*)inv_s, NGLOB / 4);
  degree_k<<<cdiv(ES_, 256), B256, 0, stream>>>(dst_s, inv_s, ES_);
  invert_k<<<cdiv(NGLOB, 256), B256, 0, stream>>>(inv_s, NGLOB);
  for (int c = 0; c < 2; ++c) {
    fill_zero4_k<<<cdiv(NCLI / 4, 256), B256, 0, stream>>>((float4*)inv_c[c], NCLI / 4);
    degree_k<<<cdiv(EC_, 256), B256, 0, stream>>>(dst_c[c], inv_c[c], EC_);
    invert_k<<<cdiv(NCLI, 256), B256, 0, stream>>>(inv_c[c], NCLI);
  }

  const unsigned totS4   = (unsigned)ES_ * 16u;   // edge x 16 float4-chunks (D=64)
  const unsigned totC4   = (unsigned)EC_ * 32u;   // edge x 32 float4-chunks (D=128)
  const unsigned n4_N64  = (unsigned)NGLOB * 16u; // N*64/4
  const unsigned n4_NC128 = (unsigned)NCLI * 32u; // NC*128/4
  const int tilesS = NGLOB / 16;                  // 6250
  const int tilesC = NCLI / 16;                   // 3125

  const float* Scur = S0;
  const float* Hcur[2] = { x0, x1 };

  for (int layer = 0; layer < 2; ++layer) {
    // ---- structure SAGEConv ----
    float* Snext = (layer == 0) ? S1 : S2;
    fill_zero4_k<<<cdiv(n4_N64, 256), B256, 0, stream>>>((float4*)agg, n4_N64);
    scatter4_k<16, 4><<<cdiv(totS4, 256), B256, 0, stream>>>(
        (const float4*)Scur, src_s, dst_s, agg, totS4);
    if (layer == 0)
      sage_gemm_k<64, true ><<<cdiv(tilesS, 4), dim3(128), 0, stream>>>(
          agg, inv_s, Scur, Wl_sp[layer], bl_s[layer], Wr_sp[layer], Snext, tilesS);
    else
      sage_gemm_k<64, false><<<cdiv(tilesS, 4), dim3(128), 0, stream>>>(
          agg, inv_s, Scur, Wl_sp[layer], bl_s[layer], Wr_sp[layer], Snext, tilesS);

    // ---- client SAGEConvs (use pre-update Scur for gather) ----
    for (int c = 0; c < 2; ++c) {
      concat4_k<<<cdiv(n4_NC128, 256), B256, 0, stream>>>(
          (const float4*)Hcur[c], (const float4*)Scur, ids[c], (float4*)xcat, n4_NC128);
      fill_zero4_k<<<cdiv(n4_NC128, 256), B256, 0, stream>>>((float4*)agg, n4_NC128);
      scatter4_k<32, 5><<<cdiv(totC4, 256), B256, 0, stream>>>(
          (const float4*)xcat, src_c[c], dst_c[c], agg, totC4);
      sage_gemm_k<128, true><<<cdiv(tilesC, 4), dim3(128), 0, stream>>>(
          agg, inv_c[c], xcat, Wl_cp[c][layer], bl_c[c][layer], Wr_cp[c][layer],
          Hbuf[c][layer], tilesC);
      Hcur[c] = Hbuf[c][layer];
    }
    Scur = Snext;
  }

  // ===== heads: softmax(concat(H2, S2[ids]) @ W + b) =====
  for (int c = 0; c < 2; ++c) {
    concat4_k<<<cdiv(n4_NC128, 256), B256, 0, stream>>>(
        (const float4*)Hcur[c], (const float4*)S2, ids[c], (float4*)xcat, n4_NC128);
    head_gemm_k<<<cdiv(tilesC, 4), dim3(128), 0, stream>>>(
        xcat, Woutp[c], bout[c], outp[c], tilesC);
    softmax16_k<<<cdiv(NCLI, 256), B256, 0, stream>>>(outp[c], NCLI);
  }
}